// ItemConv_36077725286611
// MI455X (gfx1250) — compile-verified
//
#include <hip/hip_runtime.h>
#include <math.h>

#define EMB 100
#define LAYERS 3
#define EPS 1e-12f

typedef __attribute__((ext_vector_type(2))) float v2f;
typedef __attribute__((ext_vector_type(8))) float v8f;

// ---------------------------------------------------------------------------
// init: x = embedding, out = embedding (accumulator), deg = 0
// ---------------------------------------------------------------------------
__global__ void init_kernel(const float* __restrict__ emb, float* __restrict__ x,
                            float* __restrict__ out, float* __restrict__ deg,
                            int n_elems, int n_nodes) {
    int i = blockIdx.x * blockDim.x + threadIdx.x;
    if (i < n_elems) {
        float e = emb[i];
        x[i] = e;
        out[i] = e;
    }
    if (i < n_nodes) deg[i] = 0.0f;
}

// ---------------------------------------------------------------------------
// degree: deg[col[e]] += adj_val[e]
// ---------------------------------------------------------------------------
__global__ void degree_kernel(const float* __restrict__ adj_val,
                              const int* __restrict__ adj_col,
                              float* __restrict__ deg, int n_edges) {
    int e = blockIdx.x * blockDim.x + threadIdx.x;
    if (e >= n_edges) return;
    atomicAdd(&deg[adj_col[e]], adj_val[e]);
}

// ---------------------------------------------------------------------------
// vals: vals[e] = adj_val[e] / deg[col[e]]
// ---------------------------------------------------------------------------
__global__ void vals_kernel(const float* __restrict__ adj_val,
                            const int* __restrict__ adj_col,
                            const float* __restrict__ deg,
                            float* __restrict__ vals, int n_edges) {
    int e = blockIdx.x * blockDim.x + threadIdx.x;
    if (e >= n_edges) return;
    vals[e] = adj_val[e] / deg[adj_col[e]];
}

// ---------------------------------------------------------------------------
// reset the (max, sum) reduction cell
// ---------------------------------------------------------------------------
__global__ void reset_red_kernel(float* __restrict__ red) {
    red[0] = -INFINITY;
    red[1] = 0.0f;
}

__device__ inline void atomicMaxF(float* addr, float val) {
    unsigned int* ua = (unsigned int*)addr;
    unsigned int old = __float_as_uint(*addr);
    while (__uint_as_float(old) < val) {
        unsigned int assumed = old;
        old = atomicCAS(ua, assumed, __float_as_uint(val));
        if (old == assumed) break;
    }
}

// ---------------------------------------------------------------------------
// scores[n] = dot(x[n,:], w_att) + b_att ; red[0] = max over nodes
// one wave (32 lanes) per node
// ---------------------------------------------------------------------------
__global__ void score_max_kernel(const float* __restrict__ x,
                                 const float* __restrict__ w_att,
                                 const float* __restrict__ b_att,
                                 float* __restrict__ scores,
                                 float* __restrict__ red, int n_nodes) {
    __shared__ float smax[8];
    int gid = blockIdx.x * blockDim.x + threadIdx.x;
    int node = gid >> 5;
    int lane = threadIdx.x & 31;
    int wid  = threadIdx.x >> 5;
    bool valid = node < n_nodes;

    float s = 0.0f;
    if (valid) {
        const float* xr = x + (size_t)node * EMB;
        for (int f = lane; f < EMB; f += 32) s += xr[f] * w_att[f];
    }
    // wave sum-reduce (wave32)
    for (int off = 16; off > 0; off >>= 1) s += __shfl_xor(s, off, 32);
    s += b_att[0];
    if (valid && lane == 0) scores[node] = s;
    if (lane == 0) smax[wid] = valid ? s : -INFINITY;
    __syncthreads();
    if (threadIdx.x == 0) {
        float m = smax[0];
        for (int w = 1; w < 8; ++w) m = fmaxf(m, smax[w]);
        atomicMaxF(&red[0], m);
    }
}

// ---------------------------------------------------------------------------
// scores[n] = exp(scores[n] - max) ; red[1] += sum
// ---------------------------------------------------------------------------
__global__ void exp_sum_kernel(float* __restrict__ scores,
                               float* __restrict__ red, int n_nodes) {
    __shared__ float ssum[256];
    int i = blockIdx.x * blockDim.x + threadIdx.x;
    float e = 0.0f;
    if (i < n_nodes) {
        e = __expf(scores[i] - red[0]);
        scores[i] = e;
    }
    ssum[threadIdx.x] = e;
    __syncthreads();
    for (int s = 128; s > 0; s >>= 1) {
        if (threadIdx.x < s) ssum[threadIdx.x] += ssum[threadIdx.x + s];
        __syncthreads();
    }
    if (threadIdx.x == 0) atomicAdd(&red[1], ssum[0]);
}

// ---------------------------------------------------------------------------
// WMMA GEMM fused with attention weighting:
//   m_out[n, j] = (sum_k x[n,k] * W[j,k]) * (scores[n] / red[1])
// One wave computes one 16x16 tile via V_WMMA_F32_16X16X4_F32, K = 100 = 25*4.
// B columns >= EMB are handled by a clamped pointer (in-bounds load) times a
// 0/1 float mask -> unconditional global_load_b64, no EXEC-mask branches.
// ---------------------------------------------------------------------------
__global__ void gemm_attw_kernel(const float* __restrict__ x,
                                 const float* __restrict__ W,
                                 const float* __restrict__ scores,
                                 const float* __restrict__ red,
                                 float* __restrict__ m_out, int n_rows) {
    const int NTILES = (EMB + 15) / 16;                 // 7
    int wave = (blockIdx.x * blockDim.x + threadIdx.x) >> 5;
    int mtile = wave / NTILES;
    int ntile = wave % NTILES;
    int row0 = mtile * 16;
    if (row0 >= n_rows) return;                         // wave-uniform guard
    int col0 = ntile * 16;

    int lane  = threadIdx.x & 31;
    int lhalf = lane >> 4;                              // 0: K pair {0,1}, 1: {2,3}
    int lmod  = lane & 15;
    int arow  = row0 + lmod;                            // always < n_rows (16 | n_rows)
    int bcol  = col0 + lmod;
    bool bvalid = bcol < EMB;
    float bmask = bvalid ? 1.0f : 0.0f;

    const float* xr = x + (size_t)arow * EMB;
    const float* wr = W + (size_t)(bvalid ? bcol : 0) * EMB;   // clamped, in-bounds

    v8f acc = {};
    #pragma unroll 5
    for (int k0 = 0; k0 < EMB; k0 += 4) {
        int ka = k0 + lhalf * 2;                        // even -> 8B aligned
        v2f a = *(const v2f*)(xr + ka);
        v2f b = *(const v2f*)(wr + ka);
        b[0] *= bmask;
        b[1] *= bmask;
        acc = __builtin_amdgcn_wmma_f32_16x16x4_f32(
            /*neg_a=*/false, a, /*neg_b=*/false, b,
            /*c_mod=*/(short)0, acc, /*reuse_a=*/false, /*reuse_b=*/false);
    }

    float inv_sum = 1.0f / red[1];
    if (bvalid) {
        #pragma unroll
        for (int g = 0; g < 8; ++g) {
            int row = row0 + g + lhalf * 8;             // C/D VGPR layout
            float attw = scores[row] * inv_sum;
            m_out[(size_t)row * EMB + bcol] = acc[g] * attw;
        }
    }
}

// ---------------------------------------------------------------------------
// zero a buffer
// ---------------------------------------------------------------------------
__global__ void zero_kernel(float* __restrict__ p, int n) {
    int i = blockIdx.x * blockDim.x + threadIdx.x;
    if (i < n) p[i] = 0.0f;
}

// ---------------------------------------------------------------------------
// SpMM: out[row[e], :] += vals[e] * m[col[e], :]  (one wave per edge)
// ---------------------------------------------------------------------------
__global__ void spmm_kernel(const float* __restrict__ m,
                            const float* __restrict__ vals,
                            const int* __restrict__ adj_row,
                            const int* __restrict__ adj_col,
                            float* __restrict__ out, int n_edges) {
    int gid = blockIdx.x * blockDim.x + threadIdx.x;
    int e = gid >> 5;
    int lane = threadIdx.x & 31;
    if (e >= n_edges) return;
    int r = adj_row[e];
    int c = adj_col[e];
    float v = vals[e];
    const float* src = m + (size_t)c * EMB;
    float* dst = out + (size_t)r * EMB;
    for (int f = lane; f < EMB; f += 32)
        atomicAdd(&dst[f], v * src[f]);
}

// ---------------------------------------------------------------------------
// L2-normalize rows of xc, write back into x (next layer input) and
// accumulate into out. One wave per node.
// ---------------------------------------------------------------------------
__global__ void norm_add_kernel(const float* __restrict__ xc,
                                float* __restrict__ x,
                                float* __restrict__ out, int n_nodes) {
    int gid = blockIdx.x * blockDim.x + threadIdx.x;
    int node = gid >> 5;
    int lane = threadIdx.x & 31;
    if (node >= n_nodes) return;
    const float* src = xc + (size_t)node * EMB;
    float ss = 0.0f;
    for (int f = lane; f < EMB; f += 32) { float t = src[f]; ss += t * t; }
    for (int off = 16; off > 0; off >>= 1) ss += __shfl_xor(ss, off, 32);
    float norm = sqrtf(ss);
    float inv = 1.0f / fmaxf(norm, EPS);
    float* xd = x + (size_t)node * EMB;
    float* od = out + (size_t)node * EMB;
    for (int f = lane; f < EMB; f += 32) {
        float t = src[f] * inv;
        xd[f] = t;
        od[f] += t;
    }
}

// ---------------------------------------------------------------------------
// final scale by 1/(LAYERS+1)
// ---------------------------------------------------------------------------
__global__ void finalize_kernel(float* __restrict__ out, int n) {
    int i = blockIdx.x * blockDim.x + threadIdx.x;
    if (i < n) out[i] *= (1.0f / (LAYERS + 1));
}

// ---------------------------------------------------------------------------
extern "C" void kernel_launch(void* const* d_in, const int* in_sizes, int n_in,
                              void* d_out, int out_size, void* d_ws, size_t ws_size,
                              hipStream_t stream) {
    const float* emb     = (const float*)d_in[0];
    const float* adj_val = (const float*)d_in[1];
    const float* w_att   = (const float*)d_in[2];
    const float* b_att   = (const float*)d_in[3];
    const float* w_items = (const float*)d_in[4];
    const int*   adj_row = (const int*)d_in[5];
    const int*   adj_col = (const int*)d_in[6];
    float* out = (float*)d_out;

    const int n_nodes = in_sizes[0] / EMB;
    const int n_edges = in_sizes[1];
    const size_t feat = (size_t)n_nodes * EMB;

    // workspace layout (floats)
    float* ws     = (float*)d_ws;
    float* xa     = ws;                    // current features   [N*EMB]
    float* xb     = xa + feat;             // GEMM output (m)    [N*EMB]
    float* xc     = xb + feat;             // SpMM output        [N*EMB]
    float* deg    = xc + feat;             // [N]
    float* vals   = deg + n_nodes;         // [E]
    float* scores = vals + n_edges;        // [N]
    float* red    = scores + n_nodes;      // [2]: max, sum

    const int T = 256;
    const int g_feat  = (int)((feat + T - 1) / T);
    const int g_edge  = (n_edges + T - 1) / T;
    const int g_node  = (n_nodes + T - 1) / T;
    const int g_nwave = (int)(((size_t)n_nodes * 32 + T - 1) / T);
    const int g_ewave = (int)(((size_t)n_edges * 32 + T - 1) / T);
    const int mtiles  = (n_nodes + 15) / 16;
    const int ntiles  = (EMB + 15) / 16;
    const int g_gemm  = (int)(((size_t)mtiles * ntiles * 32 + T - 1) / T);

    init_kernel<<<g_feat, T, 0, stream>>>(emb, xa, out, deg, (int)feat, n_nodes);
    degree_kernel<<<g_edge, T, 0, stream>>>(adj_val, adj_col, deg, n_edges);
    vals_kernel<<<g_edge, T, 0, stream>>>(adj_val, adj_col, deg, vals, n_edges);

    for (int layer = 0; layer < LAYERS; ++layer) {
        const float* W = w_items + (size_t)layer * EMB * EMB;
        reset_red_kernel<<<1, 1, 0, stream>>>(red);
        score_max_kernel<<<g_nwave, T, 0, stream>>>(xa, w_att, b_att, scores, red, n_nodes);
        exp_sum_kernel<<<g_node, T, 0, stream>>>(scores, red, n_nodes);
        gemm_attw_kernel<<<g_gemm, T, 0, stream>>>(xa, W, scores, red, xb, n_nodes);
        zero_kernel<<<g_feat, T, 0, stream>>>(xc, (int)feat);
        spmm_kernel<<<g_ewave, T, 0, stream>>>(xb, vals, adj_row, adj_col, xc, n_edges);
        norm_add_kernel<<<g_nwave, T, 0, stream>>>(xc, xa, out, n_nodes);
    }
    finalize_kernel<<<g_feat, T, 0, stream>>>(out, (int)feat);
}